// BlockDiagonalLayer_37752762532044
// MI455X (gfx1250) — compile-verified
//
#include <hip/hip_runtime.h>
#include <hip/hip_bf16.h>

typedef __attribute__((ext_vector_type(2))) float v2f;
typedef __attribute__((ext_vector_type(8))) float v8f;

// Shapes from the reference
#define BATCH 2048
#define NNET  1024
#define DIN   64
#define DOUT  64
// x / out row stride in floats: N * 64
#define ROWSTRIDE (NNET * DOUT)
// padded LDS row stride for W (64 data + 4 pad -> bank-conflict-free, 16B aligned)
#define WLD 68

__global__ __launch_bounds__(256, 2) void siren_blockdiag_wmma_f32(
    const float* __restrict__ x,       // (B, N, 64)
    const float* __restrict__ w,       // (N, 64, 64) row-major: W[n, j, i]
    const float* __restrict__ bias,    // (N, 64)
    const float* __restrict__ omega,   // (N,)
    float* __restrict__ out)           // (B, N*64)
{
    const int n    = blockIdx.x;       // network id
    const int tid  = threadIdx.x;      // 0..255
    const int lane = tid & 31;
    const int wave = tid >> 5;         // 0..7
    const int r    = lane & 15;        // row-within-tile / column index
    const int half = lane >> 4;        // 0 or 1
    const int koff = half * 2;         // K sub-offset for A/B fragments

    __shared__ float wlds[64 * WLD];

    // ---- cooperative load of W_n (64x64 f32 = 16KB) into padded LDS ----
    const float* wg = w + (size_t)n * (DOUT * DIN);
#pragma unroll
    for (int itv = 0; itv < 4; ++itv) {
        const int e = itv * 1024 + tid * 4;          // 0..4095, step 4
        const int j = e >> 6;                        // output row of W
        const int k = e & 63;                        // input col of W
        const float4 v = *(const float4*)(wg + e);   // coalesced 16B/lane
        *(float4*)(&wlds[j * WLD + k]) = v;          // 16B aligned (WLD%4==0, k%4==0)
    }
    __syncthreads();

    const float om = omega[n];                       // uniform -> scalar load
    // pre-scaled bias: sin(om*(y+b)) == sin(fma(om, y, om*b))
    float omb[4];
#pragma unroll
    for (int ct = 0; ct < 4; ++ct)
        omb[ct] = om * bias[n * DOUT + ct * 16 + r]; // matches C/D column = lane%16

    const float* xb = x + (size_t)n * DOUT + koff;
    float*       ob = out + (size_t)n * DOUT;

    // each wave: 16 row-tiles of 16 batch rows
    for (int it = 0; it < 16; ++it) {
        const int b0 = (it * 8 + wave) * 16;

        // ---- hoist all 16 A fragments (x read from HBM exactly once) ----
        const float* xr = xb + (size_t)(b0 + r) * ROWSTRIDE;
        v2f a[16];
#pragma unroll
        for (int kb = 0; kb < 16; ++kb)
            a[kb] = *(const v2f*)(xr + kb * 4);      // global_load_b64

        v8f c0 = {}, c1 = {}, c2 = {}, c3 = {};

#pragma unroll
        for (int kb = 0; kb < 16; ++kb) {
            const int kk = kb * 4 + koff;
            // B fragments: compiler hoists these LDS reads into registers
            // (loaded once per network), conflict-free ds_load_b64 otherwise
            const v2f bf0 = *(const v2f*)(&wlds[(0 * 16 + r) * WLD + kk]);
            const v2f bf1 = *(const v2f*)(&wlds[(1 * 16 + r) * WLD + kk]);
            const v2f bf2 = *(const v2f*)(&wlds[(2 * 16 + r) * WLD + kk]);
            const v2f bf3 = *(const v2f*)(&wlds[(3 * 16 + r) * WLD + kk]);
            c0 = __builtin_amdgcn_wmma_f32_16x16x4_f32(false, a[kb], false, bf0,
                                                       (short)0, c0, false, false);
            c1 = __builtin_amdgcn_wmma_f32_16x16x4_f32(false, a[kb], false, bf1,
                                                       (short)0, c1, false, false);
            c2 = __builtin_amdgcn_wmma_f32_16x16x4_f32(false, a[kb], false, bf2,
                                                       (short)0, c2, false, false);
            c3 = __builtin_amdgcn_wmma_f32_16x16x4_f32(false, a[kb], false, bf3,
                                                       (short)0, c3, false, false);
        }

        // ---- epilogue: fused omega*y + omega*bias, sin, store ----
#pragma unroll
        for (int v = 0; v < 8; ++v) {
            const int brow = b0 + v + half * 8;      // C/D layout: M = v + 8*(lane/16)
            float* orow = ob + (size_t)brow * ROWSTRIDE + r;
            orow[0]  = __sinf(fmaf(om, c0[v], omb[0]));
            orow[16] = __sinf(fmaf(om, c1[v], omb[1]));
            orow[32] = __sinf(fmaf(om, c2[v], omb[2]));
            orow[48] = __sinf(fmaf(om, c3[v], omb[3]));
        }
    }
}

extern "C" void kernel_launch(void* const* d_in, const int* in_sizes, int n_in,
                              void* d_out, int out_size, void* d_ws, size_t ws_size,
                              hipStream_t stream) {
    const float* x     = (const float*)d_in[0];
    const float* w     = (const float*)d_in[1];
    const float* bias  = (const float*)d_in[2];
    const float* omega = (const float*)d_in[3];
    float* out = (float*)d_out;
    (void)in_sizes; (void)n_in; (void)out_size; (void)d_ws; (void)ws_size;

    dim3 grid(NNET);   // one workgroup per network
    dim3 block(256);   // 8 waves (wave32)
    siren_blockdiag_wmma_f32<<<grid, block, 0, stream>>>(x, w, bias, omega, out);
}